// RASGModel_9663676416316
// MI455X (gfx1250) — compile-verified
//
#include <hip/hip_runtime.h>
#include <hip/hip_bf16.h>

#define H        128
#define LBL      32
#define RELD     64
#define ATT      64
#define NLAYERS  3
#define MAXDIST  10

typedef __attribute__((ext_vector_type(2))) float v2f;
typedef __attribute__((ext_vector_type(8))) float v8f;

// ---------------------------------------------------------------------------
// x[n, 0:128] = [emb1[nl0] | emb2[nl1] | rel_table[rel_id]]
// ---------------------------------------------------------------------------
__global__ void __launch_bounds__(H)
build_x_kernel(const int* __restrict__ node_label,
               const int* __restrict__ rel_id_p,
               const float* __restrict__ emb1,
               const float* __restrict__ emb2,
               const float* __restrict__ rel_table,
               float* __restrict__ x, int n_nodes) {
    int n = blockIdx.x;
    int f = threadIdx.x;
    if (n >= n_nodes) return;
    float v;
    if (f < LBL) {
        int l = node_label[n * 2 + 0]; l = (l > MAXDIST) ? MAXDIST : l;
        v = emb1[l * LBL + f];
    } else if (f < 2 * LBL) {
        int l = node_label[n * 2 + 1]; l = (l > MAXDIST) ? MAXDIST : l;
        v = emb2[l * LBL + (f - LBL)];
    } else {
        v = rel_table[rel_id_p[0] * RELD + (f - 2 * LBL)];
    }
    x[(long)n * H + f] = v;
}

__global__ void zero_f4_kernel(float4* __restrict__ p, long n4) {
    long i = (long)blockIdx.x * blockDim.x + threadIdx.x;
    if (i < n4) p[i] = float4{0.f, 0.f, 0.f, 0.f};
}

// ---------------------------------------------------------------------------
// One wave per edge; lane l moves features [4l, 4l+4).
// aggr[dst] += x[src] + lin_rel[edge_type]   (f32 HW atomics, L2-resident)
// ---------------------------------------------------------------------------
__global__ void __launch_bounds__(256)
edge_msg_kernel(const int* __restrict__ edge_index,
                const int* __restrict__ edge_type,
                const float* __restrict__ x,
                const float* __restrict__ lin_rel_l,
                float* __restrict__ aggr, int n_edges) {
    int wave = (int)(((long)blockIdx.x * blockDim.x + threadIdx.x) >> 5);
    int lane = threadIdx.x & 31;
    if (wave >= n_edges) return;
    int s = edge_index[wave];
    int d = edge_index[n_edges + wave];
    int t = edge_type[wave];
    int f = lane * 4;
    const float4 xs = *(const float4*)(x + (long)s * H + f);
    const float4 lr = *(const float4*)(lin_rel_l + (long)t * H + f);
    float* ap = aggr + (long)d * H + f;
    atomicAdd(ap + 0, xs.x + lr.x);
    atomicAdd(ap + 1, xs.y + lr.y);
    atomicAdd(ap + 2, xs.z + lr.z);
    atomicAdd(ap + 3, xs.w + lr.w);
}

// ---------------------------------------------------------------------------
// x = relu(aggr @ Wn + bn + x), fp32 WMMA 16x16x4, one 16x16 tile per wave.
// Block = 8 waves = 8 column tiles (full 128 cols) of one 16-row tile.
// ---------------------------------------------------------------------------
__global__ void __launch_bounds__(256)
layer_update_wmma(const float* __restrict__ aggr,
                  const float* __restrict__ Wn_l,
                  const float* __restrict__ bn_l,
                  float* __restrict__ x) {
    const int wave = threadIdx.x >> 5;      // 0..7 -> column tile
    const int lane = threadIdx.x & 31;
    const int m0   = blockIdx.x * 16;       // row tile base
    const int n0   = wave * 16;             // col tile base
    const int half = lane >> 4;             // 0: lanes 0-15, 1: lanes 16-31
    const int l16  = lane & 15;
    const int col  = n0 + l16;              // B/C/D lanes hold N

    // C init: residual + bias.  c[r] -> (row = m0 + r + 8*half, col)
    v8f c;
#pragma unroll
    for (int r = 0; r < 8; ++r) {
        int row = m0 + r + 8 * half;
        c[r] = x[(long)row * H + col] + bn_l[col];
    }

#pragma unroll
    for (int k0 = 0; k0 < H; k0 += 4) {
        // A 16x4: lane holds M=l16; VGPR0/1 = K = k0+2*half, k0+2*half+1
        const float* arow = aggr + (long)(m0 + l16) * H + (k0 + 2 * half);
        v2f a; a.x = arow[0]; a.y = arow[1];
        // B 4x16: lane holds N=col; VGPR0/1 = K = k0+2*half, k0+2*half+1
        v2f b;
        b.x = Wn_l[(k0 + 2 * half) * H + col];
        b.y = Wn_l[(k0 + 2 * half + 1) * H + col];
        c = __builtin_amdgcn_wmma_f32_16x16x4_f32(
                /*neg_a=*/false, a, /*neg_b=*/false, b,
                /*c_mod=*/(short)0, c, /*reuse_a=*/false, /*reuse_b=*/false);
    }

#pragma unroll
    for (int r = 0; r < 8; ++r) {
        int row = m0 + r + 8 * half;
        x[(long)row * H + col] = fmaxf(c[r], 0.0f);
    }
}

// Scalar tail for rows not covered by full 16-row tiles (N%16 != 0).
__global__ void __launch_bounds__(H)
layer_update_tail(const float* __restrict__ aggr,
                  const float* __restrict__ Wn_l,
                  const float* __restrict__ bn_l,
                  float* __restrict__ x, int row_start, int n_nodes) {
    int row = row_start + blockIdx.x;
    int col = threadIdx.x;
    if (row >= n_nodes) return;
    float acc = bn_l[col] + x[(long)row * H + col];
    for (int k = 0; k < H; ++k)
        acc += aggr[(long)row * H + k] * Wn_l[k * H + col];
    x[(long)row * H + col] = fmaxf(acc, 0.0f);
}

// ---------------------------------------------------------------------------
// gate[n] = leaky_relu(x[n] @ Wg1 + bg1, 0.2) @ Wg2 + bg2
// ---------------------------------------------------------------------------
__global__ void __launch_bounds__(ATT)
gate_kernel(const float* __restrict__ x, const float* __restrict__ Wg1,
            const float* __restrict__ bg1, const float* __restrict__ Wg2,
            const float* __restrict__ bg2, float* __restrict__ gate,
            int n_nodes) {
    __shared__ float xs[H];
    __shared__ float red[ATT];
    int n = blockIdx.x;
    int j = threadIdx.x;
    if (n >= n_nodes) return;
    xs[j]       = x[(long)n * H + j];
    xs[j + 64]  = x[(long)n * H + j + 64];
    __syncthreads();
    float s = 0.f;
#pragma unroll 8
    for (int f = 0; f < H; ++f) s += xs[f] * Wg1[f * ATT + j];
    s += bg1[j];
    s = (s > 0.f) ? s : 0.2f * s;
    red[j] = s * Wg2[j];
    __syncthreads();
    for (int off = 32; off > 0; off >>= 1) {
        if (j < off) red[j] += red[j + off];
        __syncthreads();
    }
    if (j == 0) gate[n] = red[0] + bg2[0];
}

// Single block: stats[0]=max(gate), stats[1]=sum(exp(gate-max)); zero z[0:128]
__global__ void __launch_bounds__(1024)
softmax_stats_kernel(const float* __restrict__ gate, float* __restrict__ stats,
                     float* __restrict__ z, int n_nodes) {
    __shared__ float red[1024];
    int t = threadIdx.x;
    float m = -3.4e38f;
    for (int n = t; n < n_nodes; n += 1024) m = fmaxf(m, gate[n]);
    red[t] = m; __syncthreads();
    for (int off = 512; off > 0; off >>= 1) {
        if (t < off) red[t] = fmaxf(red[t], red[t + off]);
        __syncthreads();
    }
    float gmax = red[0]; __syncthreads();
    float s = 0.f;
    for (int n = t; n < n_nodes; n += 1024) s += __expf(gate[n] - gmax);
    red[t] = s; __syncthreads();
    for (int off = 512; off > 0; off >>= 1) {
        if (t < off) red[t] += red[t + off];
        __syncthreads();
    }
    if (t == 0) { stats[0] = gmax; stats[1] = red[0]; }
    if (t < H) z[t] = 0.f;
}

// z[f] = sum_n softmax(gate)[n] * x[n,f]   (per-block register accum -> 1 atomic)
__global__ void __launch_bounds__(H)
pool_kernel(const float* __restrict__ x, const float* __restrict__ gate,
            const float* __restrict__ stats, float* __restrict__ z,
            int n_nodes) {
    int f = threadIdx.x;
    float gmax = stats[0];
    float inv  = 1.0f / stats[1];
    float acc = 0.f;
    for (int n = blockIdx.x; n < n_nodes; n += gridDim.x)
        acc += __expf(gate[n] - gmax) * inv * x[(long)n * H + f];
    atomicAdd(&z[f], acc);
}

// score = [z | x[h] | x[t]] @ Wf + bf
__global__ void __launch_bounds__(H)
final_kernel(const float* __restrict__ z, const float* __restrict__ x,
             const int* __restrict__ h_idx, const int* __restrict__ t_idx,
             const float* __restrict__ Wf, const float* __restrict__ bf,
             float* __restrict__ out) {
    __shared__ float red[H];
    int f = threadIdx.x;
    long h = h_idx[0], t = t_idx[0];
    red[f] = z[f] * Wf[f]
           + x[h * H + f] * Wf[H + f]
           + x[t * H + f] * Wf[2 * H + f];
    __syncthreads();
    for (int off = 64; off > 0; off >>= 1) {
        if (f < off) red[f] += red[f + off];
        __syncthreads();
    }
    if (f == 0) out[0] = red[0] + bf[0];
}

// ---------------------------------------------------------------------------
extern "C" void kernel_launch(void* const* d_in, const int* in_sizes, int n_in,
                              void* d_out, int out_size, void* d_ws, size_t ws_size,
                              hipStream_t stream) {
    const int*   node_label = (const int*)d_in[0];
    const int*   edge_index = (const int*)d_in[1];
    const int*   edge_type  = (const int*)d_in[2];
    const int*   h_idx      = (const int*)d_in[3];
    const int*   t_idx      = (const int*)d_in[4];
    const int*   rel_id     = (const int*)d_in[5];
    const float* emb1       = (const float*)d_in[6];
    const float* emb2       = (const float*)d_in[7];
    const float* rel_table  = (const float*)d_in[8];
    const float* lin_rel    = (const float*)d_in[9];
    const float* Wn         = (const float*)d_in[10];
    const float* bn         = (const float*)d_in[11];
    const float* Wg1        = (const float*)d_in[12];
    const float* bg1        = (const float*)d_in[13];
    const float* Wg2        = (const float*)d_in[14];
    const float* bg2        = (const float*)d_in[15];
    const float* Wf         = (const float*)d_in[16];
    const float* bf         = (const float*)d_in[17];

    const int n_nodes = in_sizes[0] / 2;
    const int n_edges = in_sizes[2];
    const int num_rels = in_sizes[9] / (NLAYERS * H);

    // Workspace layout (all 16B-aligned for float4 traffic)
    char* ws = (char*)d_ws;
    const size_t xh_bytes = (size_t)n_nodes * H * sizeof(float);
    float* x     = (float*)(ws);
    float* aggr  = (float*)(ws + xh_bytes);
    float* gate  = (float*)(ws + 2 * xh_bytes);
    float* z     = (float*)(ws + 2 * xh_bytes + ((size_t)n_nodes * sizeof(float) + 255 & ~(size_t)255));
    float* stats = z + H;

    build_x_kernel<<<n_nodes, H, 0, stream>>>(node_label, rel_id, emb1, emb2,
                                              rel_table, x, n_nodes);

    const int  row_tiles = n_nodes / 16;
    const int  tail_rows = n_nodes - row_tiles * 16;
    const long n4        = (long)n_nodes * H / 4;

    for (int l = 0; l < NLAYERS; ++l) {
        const float* lin_rel_l = lin_rel + (size_t)l * num_rels * H;
        const float* Wn_l      = Wn + (size_t)l * H * H;
        const float* bn_l      = bn + (size_t)l * H;

        zero_f4_kernel<<<(int)((n4 + 255) / 256), 256, 0, stream>>>((float4*)aggr, n4);
        edge_msg_kernel<<<(n_edges + 7) / 8, 256, 0, stream>>>(
            edge_index, edge_type, x, lin_rel_l, aggr, n_edges);
        if (row_tiles > 0)
            layer_update_wmma<<<row_tiles, 256, 0, stream>>>(aggr, Wn_l, bn_l, x);
        if (tail_rows > 0)
            layer_update_tail<<<tail_rows, H, 0, stream>>>(aggr, Wn_l, bn_l, x,
                                                           row_tiles * 16, n_nodes);
    }

    gate_kernel<<<n_nodes, ATT, 0, stream>>>(x, Wg1, bg1, Wg2, bg2, gate, n_nodes);
    softmax_stats_kernel<<<1, 1024, 0, stream>>>(gate, stats, z, n_nodes);
    pool_kernel<<<512, H, 0, stream>>>(x, gate, stats, z, n_nodes);
    final_kernel<<<1, H, 0, stream>>>(z, x, h_idx, t_idx, Wf, bf, (float*)d_out);
}